// FullNet_79328045957657
// MI455X (gfx1250) — compile-verified
//
#include <hip/hip_runtime.h>

#define H 960
#define HH (H*H)
#define NI 16384
#define NJ 24576
#define KSPLIT 4
#define KCH (NI / KSPLIT)   // 4096 rows per K-chunk

typedef __attribute__((ext_vector_type(2))) float v2f;
typedef __attribute__((ext_vector_type(8))) float v8f;

// ---------------- Kernel 1: guide raw map + per-block min/max ----------------
__global__ void guide_kernel(const float* __restrict__ full,
                             const float* __restrict__ Mm,
                             const float* __restrict__ p_a,
                             const float* __restrict__ p_t,
                             const float* __restrict__ bias_c,
                             const float* __restrict__ bias_p,
                             float* __restrict__ graw,
                             float* __restrict__ bmin,
                             float* __restrict__ bmax) {
  __shared__ float smin[256];
  __shared__ float smax[256];
  int tid = threadIdx.x;
  int idx = blockIdx.x * 256 + tid;
  float g = 0.0f;
  bool valid = idx < HH;
  if (valid) {
#pragma unroll
    for (int c = 0; c < 3; ++c) {
      float Msum = Mm[c * 3 + 0] + Mm[c * 3 + 1] + Mm[c * 3 + 2];
      float v = full[c * HH + idx] * Msum + bias_c[c];
#pragma unroll
      for (int i = 0; i < 16; ++i) {
        float hinge = v - p_t[c * 16 + i];
        hinge = hinge > 0.0f ? hinge : 0.0f;
        g += p_a[c * 16 + i] * hinge;
      }
    }
    g += bias_p[0];
    graw[idx] = g;
  }
  smin[tid] = valid ? g :  1e30f;
  smax[tid] = valid ? g : -1e30f;
  __syncthreads();
  for (int s = 128; s > 0; s >>= 1) {
    if (tid < s) {
      smin[tid] = fminf(smin[tid], smin[tid + s]);
      smax[tid] = fmaxf(smax[tid], smax[tid + s]);
    }
    __syncthreads();
  }
  if (tid == 0) {
    bmin[blockIdx.x] = smin[0];
    bmax[blockIdx.x] = smax[0];
  }
}

// ---------------- Kernel 2: reduce block min/max to global --------------------
__global__ void reduce_minmax(const float* __restrict__ bmin,
                              const float* __restrict__ bmax,
                              int n, float* __restrict__ mm) {
  __shared__ float smin[256];
  __shared__ float smax[256];
  int tid = threadIdx.x;
  float mn = 1e30f, mx = -1e30f;
  for (int i = tid; i < n; i += 256) {
    mn = fminf(mn, bmin[i]);
    mx = fmaxf(mx, bmax[i]);
  }
  smin[tid] = mn;
  smax[tid] = mx;
  __syncthreads();
  for (int s = 128; s > 0; s >>= 1) {
    if (tid < s) {
      smin[tid] = fminf(smin[tid], smin[tid + s]);
      smax[tid] = fmaxf(smax[tid], smax[tid + s]);
    }
    __syncthreads();
  }
  if (tid == 0) { mm[0] = smin[0]; mm[1] = smax[0]; }
}

// ---------------- Kernel 3: generic 3x3 conv (pad 1) + ReLU -------------------
__global__ void conv_relu(const float* __restrict__ in,
                          const float* __restrict__ w,
                          const float* __restrict__ b,
                          float* __restrict__ out,
                          int Cin, int Cout, int Hin, int Win,
                          int Hout, int Wout, int stride) {
  int t = blockIdx.x * blockDim.x + threadIdx.x;
  int total = Cout * Hout * Wout;
  if (t >= total) return;
  int x = t % Wout;
  int y = (t / Wout) % Hout;
  int o = t / (Wout * Hout);
  float acc = b[o];
  for (int ci = 0; ci < Cin; ++ci) {
    const float* inp = in + ci * Hin * Win;
    const float* wp = w + ((o * Cin + ci) * 9);
#pragma unroll
    for (int ky = 0; ky < 3; ++ky) {
      int iy = y * stride - 1 + ky;
      if (iy < 0 || iy >= Hin) continue;
#pragma unroll
      for (int kx = 0; kx < 3; ++kx) {
        int ix = x * stride - 1 + kx;
        if (ix < 0 || ix >= Win) continue;
        acc += inp[iy * Win + ix] * wp[ky * 3 + kx];
      }
    }
  }
  out[t] = acc > 0.0f ? acc : 0.0f;
}

// ---------------- Kernel 4: bandwidth-bound GEMV via fp32 WMMA, K-split -------
// partial[kc*NJ + j] = sum_{i in chunk kc} f[i]*W[i*NJ+j]
// A tile (16x4): A[m][k] = W[i+k][j+m]; B tile (4x16) broadcasts f[i+k].
// 6144 waves (1536 j-tiles x 4 K-chunks) for HBM latency hiding.
__global__ void gemv_wmma(const float* __restrict__ f,
                          const float* __restrict__ W,
                          float* __restrict__ partial) {
  int gwave = (blockIdx.x * blockDim.x + threadIdx.x) >> 5;
  int lane = threadIdx.x & 31;
  const int ntiles = NJ / 16;            // 1536
  int jt = gwave % ntiles;
  int kc = gwave / ntiles;               // 0..KSPLIT-1
  if (kc >= KSPLIT) return;
  int j = jt * 16;
  int half = lane >> 4;                  // 0: lanes 0-15 (K=0,1); 1: lanes 16-31 (K=2,3)
  int col  = lane & 15;
  int r0   = half * 2;
  int i0   = kc * KCH;
  int iend = i0 + KCH;

  v8f acc0 = {}, acc1 = {}, acc2 = {}, acc3 = {};
  for (int i = i0; i < iend; i += 16) {
    // speculative prefetch 64 rows ahead
    if (i + 64 < iend)
      __builtin_prefetch(&W[(size_t)(i + 64 + r0) * NJ + j + col], 0, 1);
#pragma unroll
    for (int u = 0; u < 4; ++u) {
      int ii = i + 4 * u;
      v2f a, bvec;
      a.x    = W[(size_t)(ii + r0)     * NJ + j + col];
      a.y    = W[(size_t)(ii + r0 + 1) * NJ + j + col];
      bvec.x = f[ii + r0];
      bvec.y = f[ii + r0 + 1];
      if (u == 0)
        acc0 = __builtin_amdgcn_wmma_f32_16x16x4_f32(false, a, false, bvec,
                                                     (short)0, acc0, false, false);
      else if (u == 1)
        acc1 = __builtin_amdgcn_wmma_f32_16x16x4_f32(false, a, false, bvec,
                                                     (short)0, acc1, false, false);
      else if (u == 2)
        acc2 = __builtin_amdgcn_wmma_f32_16x16x4_f32(false, a, false, bvec,
                                                     (short)0, acc2, false, false);
      else
        acc3 = __builtin_amdgcn_wmma_f32_16x16x4_f32(false, a, false, bvec,
                                                     (short)0, acc3, false, false);
    }
  }
  v8f acc = acc0 + acc1 + acc2 + acc3;
  // D layout: VGPR r, lanes 0-15 -> M=r; lanes 16-31 -> M=8+r.
  if (col == 0) {
    int base = j + half * 8;
#pragma unroll
    for (int m = 0; m < 8; ++m)
      partial[kc * NJ + base + m] = acc[m];
  }
}

// ---------------- Kernel 4b: sum K-chunk partials + bias + ReLU ---------------
__global__ void gemv_finish(const float* __restrict__ partial,
                            const float* __restrict__ Ab,
                            float* __restrict__ Aout) {
  int j = blockIdx.x * 256 + threadIdx.x;
  if (j >= NJ) return;
  float v = ((partial[j] + partial[NJ + j]) +
             (partial[2 * NJ + j] + partial[3 * NJ + j])) + Ab[j];
  Aout[j] = v > 0.0f ? v : 0.0f;
}

// ---------------- Kernel 5: trilinear slice (replicating reference quirks) ----
__global__ void apply_kernel(const float* __restrict__ full,
                             const float* __restrict__ graw,
                             const float* __restrict__ mm,
                             const float* __restrict__ Atab,
                             float* __restrict__ out) {
  int idx = blockIdx.x * 256 + threadIdx.x;
  if (idx >= HH) return;
  int h = idx / H;
  int w = idx % H;
  float gmin = mm[0], gmax = mm[1];
  float g = (graw[idx] - gmin) / (gmax - gmin);

  int x0 = h >> 6, x1 = x0 + 1;
  float xd = (float)(h & 63) * (1.0f / 64.0f);
  int y0 = w >> 6, y1 = y0 + 1;
  float yd = (float)(w & 63) * (1.0f / 64.0f);
  float z = g * 6.0f;
  int z0 = (int)floorf(z);
  float zd = z - (float)z0;
  int z1 = z0 + 1;

  int b000 = ((x0 * 16 + y0) * 8 + z0) * 12;
  int b100 = ((x1 * 16 + y0) * 8 + z0) * 12;
  int b010 = ((x0 * 16 + y1) * 8 + z0) * 12;
  int b110 = ((x1 * 16 + y1) * 8 + z0) * 12;
  int b001 = ((x0 * 16 + y0) * 8 + z1) * 12;
  int b101 = ((x1 * 16 + y0) * 8 + z1) * 12;
  int b011 = ((x0 * 16 + y1) * 8 + z1) * 12;
  int b111 = ((x1 * 16 + y1) * 8 + z1) * 12;

  float Ares[12];
#pragma unroll
  for (int o = 0; o < 12; ++o) {
    float a000 = Atab[b000 + o], a100 = Atab[b100 + o];
    float a010 = Atab[b010 + o], a110 = Atab[b110 + o];
    float a001 = Atab[b001 + o], a101 = Atab[b101 + o];
    float a011 = Atab[b011 + o], a111 = Atab[b111 + o];
    // reference: p = A[X0]*xd + A[X1]*(1-xd)  (note the "backwards" weighting)
    float p1 = a000 * xd + a100 * (1.0f - xd);
    float p2 = a010 * xd + a110 * (1.0f - xd);
    float p3 = a001 * xd + a101 * (1.0f - xd);
    float p4 = a011 * xd + a111 * (1.0f - xd);
    float q1 = p1 * yd + p2 * (1.0f - yd);
    float q2 = p3 * yd + p4 * (1.0f - yd);
    Ares[o] = q1 * zd + q2 * (1.0f - yd);  // reference bug: (1-ydb), not (1-zdb)
  }
#pragma unroll
  for (int c = 0; c < 3; ++c) {
    float scale = Ares[c * 4 + 0] + Ares[c * 4 + 1] + Ares[c * 4 + 2];
    float off = Ares[c * 4 + 3];
    out[c * HH + idx] = full[c * HH + idx] * scale + off;
  }
}

extern "C" void kernel_launch(void* const* d_in, const int* in_sizes, int n_in,
                              void* d_out, int out_size, void* d_ws, size_t ws_size,
                              hipStream_t stream) {
  (void)in_sizes; (void)n_in; (void)out_size; (void)ws_size;
  const float* full   = (const float*)d_in[0];
  const float* low    = (const float*)d_in[1];
  const float* Mm     = (const float*)d_in[2];
  const float* p_a    = (const float*)d_in[3];
  const float* p_t    = (const float*)d_in[4];
  const float* bias_c = (const float*)d_in[5];
  const float* bias_p = (const float*)d_in[6];
  const float* s1_w = (const float*)d_in[7];  const float* s1_b = (const float*)d_in[8];
  const float* s2_w = (const float*)d_in[9];  const float* s2_b = (const float*)d_in[10];
  const float* s3_w = (const float*)d_in[11]; const float* s3_b = (const float*)d_in[12];
  const float* s4_w = (const float*)d_in[13]; const float* s4_b = (const float*)d_in[14];
  const float* l1_w = (const float*)d_in[15]; const float* l1_b = (const float*)d_in[16];
  const float* l2_w = (const float*)d_in[17]; const float* l2_b = (const float*)d_in[18];
  const float* A_w  = (const float*)d_in[19]; const float* A_b  = (const float*)d_in[20];
  float* out = (float*)d_out;
  float* ws = (float*)d_ws;

  float* graw = ws;                       // 921600
  float* bmin = graw + HH;                // 3600
  float* bmax = bmin + 3600;              // 3600
  float* mm   = bmax + 3600;              // 2 (+2 pad)
  float* s1   = mm + 4;                   // 8*128*128  = 131072
  float* s2   = s1 + 8 * 128 * 128;       // 16*64*64   = 65536
  float* s3   = s2 + 16 * 64 * 64;        // 32*32*32   = 32768
  float* s4   = s3 + 32 * 32 * 32;        // 64*16*16   = 16384
  float* l1o  = s4 + 64 * 16 * 16;        // 16384
  float* l2o  = l1o + 64 * 16 * 16;       // 16384
  float* part = l2o + 64 * 16 * 16;       // KSPLIT*NJ  = 98304
  float* Aout = part + KSPLIT * NJ;       // 24576
  // total ~ 1.33M floats (~5.3 MB) of workspace

  guide_kernel<<<HH / 256, 256, 0, stream>>>(full, Mm, p_a, p_t, bias_c, bias_p,
                                             graw, bmin, bmax);
  reduce_minmax<<<1, 256, 0, stream>>>(bmin, bmax, HH / 256, mm);

  conv_relu<<<(8 * 128 * 128 + 255) / 256, 256, 0, stream>>>(
      low, s1_w, s1_b, s1, 3, 8, 256, 256, 128, 128, 2);
  conv_relu<<<(16 * 64 * 64 + 255) / 256, 256, 0, stream>>>(
      s1, s2_w, s2_b, s2, 8, 16, 128, 128, 64, 64, 2);
  conv_relu<<<(32 * 32 * 32 + 255) / 256, 256, 0, stream>>>(
      s2, s3_w, s3_b, s3, 16, 32, 64, 64, 32, 32, 2);
  conv_relu<<<(64 * 16 * 16 + 255) / 256, 256, 0, stream>>>(
      s3, s4_w, s4_b, s4, 32, 64, 32, 32, 16, 16, 2);
  conv_relu<<<(64 * 16 * 16 + 255) / 256, 256, 0, stream>>>(
      s4, l1_w, l1_b, l1o, 64, 64, 16, 16, 16, 16, 1);
  conv_relu<<<(64 * 16 * 16 + 255) / 256, 256, 0, stream>>>(
      l1o, l2_w, l2_b, l2o, 64, 64, 16, 16, 16, 16, 1);

  // 6144 waves (1536 j-tiles x 4 K-chunks), 8 waves (256 threads) per block
  gemv_wmma<<<(NJ / 16) * KSPLIT / 8, 256, 0, stream>>>(l2o, A_w, part);
  gemv_finish<<<(NJ + 255) / 256, 256, 0, stream>>>(part, A_b, Aout);

  apply_kernel<<<HH / 256, 256, 0, stream>>>(full, graw, mm, Aout, out);
}